// DeepPolyAffineTransformer_20452634263557
// MI455X (gfx1250) — compile-verified
//
#include <hip/hip_runtime.h>

// ---------------------------------------------------------------------------
// DeepPoly affine bound propagation, restructured as a 6-column GEMV bundle
// executed with V_WMMA_F32_16X16X4_F32 on gfx1250 (wave32). The 128 MB weight
// stream is staged into LDS by the CDNA5 Tensor Data Mover (TENSOR_LOAD_TO_LDS,
// TENSORcnt-tracked, one descriptor per 16x64 tile, hardware LDS padding),
// triple-buffered, with zero workgroup barriers (each wave owns its stripe).
//
//  lower      = W@c1 + |W|@c2 + b          c1=(l+u)/2   c2=(l-u)/2
//  upper      = W@c1 - |W|@c2 + b
//  new_lower  = W@c3 + |W|@c4 + b + (W@(mu/2) - |W|@(mu/2))
//  new_upper  = W@c3 - |W|@c4 + b + (W@(mu/2) + |W|@(mu/2))
//  out        = [max(lower,new_lower), min(upper,new_upper)]
//  where x=beta*l0, y=lmbda*u0, c3=(x+y)/2, c4=(x-y)/2
//  (valid because beta,lmbda >= 0, so the ReLU-relaxation clamps factor into
//   Wp*beta / Wm*lmbda exactly).
// ---------------------------------------------------------------------------

typedef __attribute__((ext_vector_type(2))) float    v2f;
typedef __attribute__((ext_vector_type(4))) float    v4f;
typedef __attribute__((ext_vector_type(8))) float    v8f;
typedef __attribute__((ext_vector_type(4))) unsigned v4u;
typedef __attribute__((ext_vector_type(8))) unsigned v8u;

#define N_OUT 4096
#define N_IN  8192

#define KSLICES 8
#define KSLICE_LEN (N_IN / KSLICES)          // 1024
#define ROWS_PER_WG 64
#define WAVES_PER_WG 4
#define THREADS_PER_WG (WAVES_PER_WG * 32)   // 128 (gfx1250 is wave32)
#define KC 64                                 // k-chunk staged in LDS
#define NBUF 3                                // triple buffer (2 TDM in flight)
#define LDS_PITCH 68                          // dwords/row; pad so 16-row b64
                                              // A-fragment reads hit 64 banks

// ws layout
#define BPACK_FLOATS ((N_IN / 4) * 32 * 4)            // 262144 floats = 1 MB
#define PARTIALS_FLOATS (KSLICES * N_OUT * 8)         // 262144 floats = 1 MB

// ---------------------------------------------------------------------------
// Kernel 1: pack B panels [K x 16] for the W-GEMM and the |W|-GEMM in the
// native WMMA B-operand lane layout:
//   lane L (0..31), k-step t: kk = 4t + 2*(L>=16), col c = L&15
//   v0 = B[kk,   c], v1 = B[kk+1, c]
// W-panel cols:  {c1, c3, mu/2, 0...};  |W|-panel cols: {c2, c4, mu/2, 0...}
// Stored interleaved per lane as float4 {bw0,bw1,ba0,ba1} -> one b128 load.
// ---------------------------------------------------------------------------
__global__ void deeppoly_build_bpack(const float* __restrict__ bounds,
                                     const float* __restrict__ bounds0,
                                     const float* __restrict__ beta,
                                     const float* __restrict__ lmbda,
                                     const float* __restrict__ mu,
                                     float* __restrict__ bpack) {
  int gid = blockIdx.x * blockDim.x + threadIdx.x;   // 0 .. (N_IN/4)*32-1
  int t   = gid >> 5;
  int L   = gid & 31;
  int c   = L & 15;
  int kk  = t * 4 + ((L >> 4) << 1);

  float v0 = 0.f, v1 = 0.f, v2 = 0.f, v3 = 0.f;
  if (c < 3) {
    float vw[2], va[2];
#pragma unroll
    for (int j = 0; j < 2; ++j) {
      int k = kk + j;
      if (c == 0) {
        float l = bounds[k], u = bounds[N_IN + k];
        vw[j] = 0.5f * (l + u);
        va[j] = 0.5f * (l - u);
      } else if (c == 1) {
        float x = beta[k]  * bounds0[k];
        float y = lmbda[k] * bounds0[N_IN + k];
        vw[j] = 0.5f * (x + y);
        va[j] = 0.5f * (x - y);
      } else {
        vw[j] = 0.5f * mu[k];
        va[j] = vw[j];
      }
    }
    v0 = vw[0]; v1 = vw[1]; v2 = va[0]; v3 = va[1];
  }
  v4f o = {v0, v1, v2, v3};
  *(v4f*)(bpack + (size_t)gid * 4) = o;
}

// ---------------------------------------------------------------------------
// Kernel 2: streaming GEMM. Each wave owns a 16-row WMMA tile and TDM-loads
// its own 16x64 f32 stripe per chunk (no cross-wave LDS sharing -> no
// barriers). Triple-buffered LDS, 2 TDM descriptors in flight per wave.
// ---------------------------------------------------------------------------
__global__ __launch_bounds__(THREADS_PER_WG)
void deeppoly_wmma_gemm(const float* __restrict__ W,
                        const float* __restrict__ bpack,
                        float* __restrict__ partials) {
  __shared__ __align__(16) float lds[NBUF][ROWS_PER_WG * LDS_PITCH];

  const int tid  = threadIdx.x;
  const int lane = tid & 31;
  const int wave = tid >> 5;
  const int mtile = blockIdx.x;                 // 0..63
  const int slice = blockIdx.y;                 // 0..KSLICES-1
  const int rowBase = mtile * ROWS_PER_WG;
  const int k0slice = slice * KSLICE_LEN;

  // byte offset of this wave's 16-row stripe in each LDS buffer
  // (generic LDS pointers keep the workgroup-relative offset in bits [31:0])
  unsigned stripeByte[NBUF];
#pragma unroll
  for (int b = 0; b < NBUF; ++b)
    stripeByte[b] =
        (unsigned)(size_t)(void*)&lds[b][wave * 16 * LDS_PITCH];

  // --- TDM descriptor group 1 (invariant) ---------------------------------
  // d0: workgroup_mask=0 | data_size=4B(2)<<16 | pad_enable<<20 |
  //     pad_interval=64dw(5)<<22 | pad_amount=4dw(3)<<25
  // tensor_dim0=8192, tensor_dim1=4096, tile_dim0=64, tile_dim1=16,
  // tile_dim2=0 (2D), tensor_dim0_stride=8192, dim1 stride unused.
  v8u g1;
  g1[0] = (2u << 16) | (1u << 20) | (5u << 22) | (3u << 25);
  g1[1] = (unsigned)((N_IN & 0xFFFFu) << 16);                  // dim0 lo16
  g1[2] = (unsigned)((N_IN >> 16) | ((N_OUT & 0xFFFFu) << 16));// dim0 hi|dim1 lo
  g1[3] = (unsigned)((N_OUT >> 16) | (KC << 16));              // dim1 hi|tile0
  g1[4] = 16u;                                                 // tile1 | tile2=0
  g1[5] = (unsigned)N_IN;                                      // stride0 lo32
  g1[6] = 0u;                                                  // stride0 hi16|..
  g1[7] = 0u;

  // global byte address of this wave's tile at chunk 0
  const unsigned long long gbase = (unsigned long long)(const void*)(
      W + (size_t)(rowBase + wave * 16) * N_IN + k0slice);

  auto tdm_issue = [&](int buf, int chunk) {
    unsigned long long ga = gbase + (unsigned long long)chunk * (KC * 4);
    v4u g0;
    g0[0] = 1u;                                   // count=1, user mode
    g0[1] = stripeByte[buf];                      // lds_addr (bytes)
    g0[2] = (unsigned)ga;                         // global_addr[31:0]
    g0[3] = (unsigned)((ga >> 32) & 0x01FFFFFFull) | (2u << 30); // [56:32]|type=2
    // dscnt 0: all my prior ds reads of this buffer have retired (WAR-safe)
    asm volatile("s_wait_dscnt 0x0\n\t"
                 "tensor_load_to_lds %0, %1"
                 :: "s"(g0), "s"(g1)
                 : "memory");
  };

  // A-fragment (16x4 f32) lane mapping: lane<16 -> row=lane, k=+0,+1
  //                                     lane>=16 -> row=lane-16, k=+2,+3
  const int aRow  = wave * 16 + (lane & 15);
  const int aKoff = (lane >> 4) * 2;

  v8f accw = {0.f, 0.f, 0.f, 0.f, 0.f, 0.f, 0.f, 0.f};
  v8f acca = {0.f, 0.f, 0.f, 0.f, 0.f, 0.f, 0.f, 0.f};

  const int nchunks = KSLICE_LEN / KC;            // 16
  tdm_issue(0, 0);
  if (nchunks > 1) tdm_issue(1, 1);

  for (int ch = 0; ch < nchunks; ++ch) {
    const int buf = ch % NBUF;
    if (ch + 2 < nchunks) tdm_issue((ch + 2) % NBUF, ch + 2);

    // TDM ops complete in order: waiting down to the number of still-wanted
    // younger descriptors guarantees chunk ch has landed in LDS.
    const int rem = nchunks - 1 - ch;
    if (rem >= 2)      asm volatile("s_wait_tensorcnt 2" ::: "memory");
    else if (rem == 1) asm volatile("s_wait_tensorcnt 1" ::: "memory");
    else               asm volatile("s_wait_tensorcnt 0" ::: "memory");

    const float* lbase = &lds[buf][aRow * LDS_PITCH + aKoff];
    const float* bbase = bpack +
        ((size_t)(k0slice / 4 + ch * (KC / 4)) * 32 + lane) * 4;

#pragma unroll
    for (int t = 0; t < KC / 4; ++t) {
      v2f aw = *(const v2f*)(lbase + t * 4);              // ds_load_b64
      v2f aa;
      aa.x = __builtin_fabsf(aw.x);
      aa.y = __builtin_fabsf(aw.y);
      v4f bq = *(const v4f*)(bbase + (size_t)t * 128);    // b128, L2-resident
      v2f bw = {bq.x, bq.y};
      v2f ba = {bq.z, bq.w};
      accw = __builtin_amdgcn_wmma_f32_16x16x4_f32(
          false, aw, false, bw, (short)0, accw, false, false);
      acca = __builtin_amdgcn_wmma_f32_16x16x4_f32(
          false, aa, false, ba, (short)0, acca, false, false);
    }
  }

  // C/D layout: vgpr r -> row r (lanes 0-15, col=lane) / row r+8 (lanes 16-31,
  // col=lane-16). Only columns 0..2 carry data.
  const int col = lane & 15;
  if (col < 3) {
    const int rhalf = (lane >> 4) * 8;
#pragma unroll
    for (int r = 0; r < 8; ++r) {
      int row = rowBase + wave * 16 + rhalf + r;
      float* p = partials + ((size_t)slice * N_OUT + row) * 8 + col * 2;
      p[0] = accw[r];
      p[1] = acca[r];
    }
  }
}

// ---------------------------------------------------------------------------
// Kernel 3: reduce K-slices, apply bias, form bounds, keep tighter.
// ---------------------------------------------------------------------------
__global__ void deeppoly_finalize(const float* __restrict__ partials,
                                  const float* __restrict__ bias,
                                  float* __restrict__ out) {
  int row = blockIdx.x * blockDim.x + threadIdx.x;
  if (row >= N_OUT) return;
  float Sw0 = 0.f, Sa0 = 0.f, Sw1 = 0.f, Sa1 = 0.f, Sw2 = 0.f, Sa2 = 0.f;
#pragma unroll
  for (int s = 0; s < KSLICES; ++s) {
    const float* p = partials + ((size_t)s * N_OUT + row) * 8;
    Sw0 += p[0]; Sa0 += p[1];
    Sw1 += p[2]; Sa1 += p[3];
    Sw2 += p[4]; Sa2 += p[5];
  }
  float b  = bias[row];
  float lf = Sw0 + Sa0 + b;            // forward lower
  float uf = Sw0 - Sa0 + b;            // forward upper
  float wm_mu = Sw2 - Sa2;             // Wm @ mu
  float wp_mu = Sw2 + Sa2;             // Wp @ mu
  float nl = Sw1 + Sa1 + b + wm_mu;    // back-substituted lower
  float nu = Sw1 - Sa1 + b + wp_mu;    // back-substituted upper
  out[row]          = fmaxf(lf, nl);
  out[N_OUT + row]  = fminf(uf, nu);
}

// ---------------------------------------------------------------------------
extern "C" void kernel_launch(void* const* d_in, const int* in_sizes, int n_in,
                              void* d_out, int out_size, void* d_ws,
                              size_t ws_size, hipStream_t stream) {
  const float* weight  = (const float*)d_in[0];
  const float* bias    = (const float*)d_in[1];
  const float* bounds  = (const float*)d_in[2];
  const float* bounds0 = (const float*)d_in[3];
  const float* beta    = (const float*)d_in[4];
  const float* lmbda   = (const float*)d_in[5];
  const float* mu      = (const float*)d_in[6];
  float* out = (float*)d_out;

  float* bpack    = (float*)d_ws;                       // 1 MB
  float* partials = bpack + BPACK_FLOATS;               // 1 MB

  deeppoly_build_bpack<<<(N_IN / 4) * 32 / 256, 256, 0, stream>>>(
      bounds, bounds0, beta, lmbda, mu, bpack);

  dim3 grid(N_OUT / ROWS_PER_WG, KSLICES);
  deeppoly_wmma_gemm<<<grid, THREADS_PER_WG, 0, stream>>>(
      weight, bpack, partials);

  deeppoly_finalize<<<(N_OUT + 255) / 256, 256, 0, stream>>>(
      partials, bias, out);
}